// MinibatchDiscrimination_45286135169752
// MI455X (gfx1250) — compile-verified
//
#include <hip/hip_runtime.h>

// Problem constants (from the reference)
#define NROWS 512      // N
#define INF   1024     // IN_FEATURES
#define BF    64       // OUT_FEATURES
#define KD    16       // KERNEL_DIMS
#define MK    (BF*KD)  // 1024 = columns of the projection matrix
#define OUTF  (INF+BF) // 1088 output columns

typedef __attribute__((ext_vector_type(2))) float v2f;
typedef __attribute__((ext_vector_type(8))) float v8f;

// ---------------------------------------------------------------------------
// Kernel 1: m[512,1024] = x[512,1024] @ W[1024,1024]  (W = T reshaped)
// One wave32 per 16x16 output tile, fp32 WMMA (exact fp32 semantics vs the
// reference), K-loop in steps of 4.
//
// V_WMMA_F32_16X16X4_F32 operand layouts (ISA 7.12.2):
//   A (16x4, MxK): lane = M + 16*(K>=2), VGPR = K%2
//   B (4x16, KxN): lane = N + 16*(K>=2), VGPR = K%2   (rows striped over lanes)
//   C/D (16x16):   VGPR r holds row (lane<16 ? r : 8+r), col = lane&15
// ---------------------------------------------------------------------------
__global__ __launch_bounds__(32) void mbd_proj_wmma(
    const float* __restrict__ x,   // [512,1024]
    const float* __restrict__ w,   // [1024,1024]
    float* __restrict__ m)         // [512,1024]
{
  const int tn   = blockIdx.x;        // 0..63  (N tile)
  const int tm   = blockIdx.y;        // 0..31  (M tile)
  const int lane = threadIdx.x;       // 0..31
  const int half = lane >> 4;         // 0 or 1 -> K-halves {0,1} vs {2,3}
  const int l16  = lane & 15;

  const int arow = tm * 16 + l16;     // A: M index carried by lane&15
  const int bcol = tn * 16 + l16;     // B: N index carried by lane&15

  const float* aptr = x + (size_t)arow * INF + 2 * half;     // K = 2*half + {0,1}
  const float* bptr = w + (size_t)(2 * half) * MK + bcol;

  v8f c = {0.f, 0.f, 0.f, 0.f, 0.f, 0.f, 0.f, 0.f};

  for (int kk = 0; kk < INF; kk += 4) {
    v2f a;
    a.x = aptr[kk];                   // A[m][kk + 2*half + 0]
    a.y = aptr[kk + 1];               // A[m][kk + 2*half + 1]
    v2f b;
    b.x = bptr[(size_t)kk * MK];            // B[kk + 2*half + 0][n]
    b.y = bptr[(size_t)(kk + 1) * MK];      // B[kk + 2*half + 1][n]
    // 8 args: (neg_a, A, neg_b, B, c_mod, C, reuse_a, reuse_b)
    c = __builtin_amdgcn_wmma_f32_16x16x4_f32(false, a, false, b,
                                              (short)0, c, false, false);
  }

  // Scatter C: VGPR r -> row 8*half + r, col = lane&15
  float* mout = m + (size_t)(tm * 16 + half * 8) * MK + tn * 16 + l16;
#pragma unroll
  for (int r = 0; r < 8; ++r) {
    mout[(size_t)r * MK] = c[r];
  }
}

// ---------------------------------------------------------------------------
// Kernel 2: features[i,b] = sum_j exp(-sum_k |m[i,b,k] - m[j,b,k]|)
// Block = 256 threads = 4 i-rows x 64 features. Stage 8 j-rows (32 KB) of m
// into LDS per iteration; every thread consumes all 8 staged rows.
// Prefetch the next chunk (global_prefetch_b8) while consuming the current
// one from LDS to hide L2 latency.
// ---------------------------------------------------------------------------
#define ITILE  4
#define JCHUNK 8

__global__ __launch_bounds__(256) void mbd_features(
    const float* __restrict__ m,   // [512,1024]  (= [512][64][16])
    float* __restrict__ out)       // [512,1088]
{
  __shared__ float smj[JCHUNK * MK];      // 8 * 1024 * 4B = 32 KB

  const int tid = threadIdx.x;            // 0..255
  const int b   = tid & (BF - 1);         // 0..63  feature index
  const int il  = tid >> 6;               // 0..3   local i
  const int i   = blockIdx.x * ITILE + il;

  // This thread's own 16-vector m[i,b,:]
  float mi[KD];
  const float* mip = m + (size_t)i * MK + b * KD;
#pragma unroll
  for (int k = 0; k < KD; ++k) mi[k] = mip[k];

  float acc = 0.f;
  for (int j0 = 0; j0 < NROWS; j0 += JCHUNK) {
    __syncthreads();  // protect previous chunk's readers
    // Cooperative load: 8192 floats = 2048 float4, 256 threads -> 8 each
    const float4* src = (const float4*)(m + (size_t)j0 * MK);
    float4* dst = (float4*)smj;
    for (int t = tid; t < (JCHUNK * MK) / 4; t += 256) dst[t] = src[t];

    // Prefetch next chunk into cache (gfx1250 global_prefetch_b8); each
    // thread touches one 128B-spaced address covering the 32KB chunk.
    if (j0 + JCHUNK < NROWS) {
      const char* nxt = (const char*)(m + (size_t)(j0 + JCHUNK) * MK);
      __builtin_prefetch(nxt + (size_t)tid * 128, /*rw=*/0, /*locality=*/3);
    }
    __syncthreads();

#pragma unroll
    for (int jj = 0; jj < JCHUNK; ++jj) {
      const float* mj = smj + jj * MK + b * KD;
      float l1 = 0.f;
#pragma unroll
      for (int k = 0; k < KD; ++k) l1 += fabsf(mj[k] - mi[k]);
      acc += __expf(-l1);
    }
  }

  out[(size_t)i * OUTF + INF + b] = acc;
}

// ---------------------------------------------------------------------------
// Kernel 3: copy x into out[:, 0:1024] (float4 wide; 1088 cols = 272 float4)
// ---------------------------------------------------------------------------
__global__ __launch_bounds__(256) void mbd_copy_x(
    const float* __restrict__ x, float* __restrict__ out)
{
  int idx = blockIdx.x * 256 + threadIdx.x;   // over 512*256 float4's
  int i    = idx >> 8;                        // row
  int col4 = idx & 255;                       // float4 column
  ((float4*)out)[(size_t)i * (OUTF / 4) + col4] = ((const float4*)x)[idx];
}

// ---------------------------------------------------------------------------
extern "C" void kernel_launch(void* const* d_in, const int* in_sizes, int n_in,
                              void* d_out, int out_size, void* d_ws, size_t ws_size,
                              hipStream_t stream) {
  (void)in_sizes; (void)n_in; (void)out_size; (void)ws_size;

  const float* x = (const float*)d_in[0];   // [512,1024] fp32
  const float* T = (const float*)d_in[1];   // [1024,64,16] fp32 == W [1024,1024]
  float* out = (float*)d_out;               // [512,1088] fp32
  float* m   = (float*)d_ws;                // [512,1024] fp32 scratch (2 MB)

  // GEMM: 64 x 32 tiles of 16x16, one wave each
  mbd_proj_wmma<<<dim3(64, 32), dim3(32), 0, stream>>>(x, T, m);

  // Features: 128 blocks x (4 rows * 64 features)
  mbd_features<<<dim3(NROWS / ITILE), dim3(256), 0, stream>>>(m, out);

  // Concat x into the output
  mbd_copy_x<<<dim3((NROWS * INF / 4) / 256), dim3(256), 0, stream>>>(x, out);
}